// TTTLinear_37452114821646
// MI455X (gfx1250) — compile-verified
//
#include <hip/hip_runtime.h>
#include <hip/hip_bf16.h>
#include <math.h>

// ---------------------------------------------------------------------------
// TTT-Linear forward for MI455X (gfx1250), fp32 throughout via
// V_WMMA_F32_16X16X4_F32 (wave32), with TDM (tensor_load_to_lds) double
// buffering of the scan's chunk feeds and wave32 shfl-parallel LayerNorms.
// ---------------------------------------------------------------------------

typedef float v2f __attribute__((ext_vector_type(2)));
typedef float v8f __attribute__((ext_vector_type(8)));
typedef unsigned int u32x4 __attribute__((ext_vector_type(4)));
typedef int i32x4 __attribute__((ext_vector_type(4)));
typedef int i32x8 __attribute__((ext_vector_type(8)));

#define WMMA_F32X4(Af, Bf, Cf) \
  __builtin_amdgcn_wmma_f32_16x16x4_f32(false, (Af), false, (Bf), (short)0, (Cf), false, false)

namespace {
constexpr int Bb  = 4;
constexpr int Nn  = 4096;
constexpr int Dd  = 1024;
constexpr int NHh = 16;
constexpr int HFf = 64;
constexpr int Mc  = 64;     // chunk length
constexpr int Kc  = 4;      // conv taps
constexpr int COLS = 3 * Dd + NHh;   // 3088
}

// ---------------------------------------------------------------------------
// TDM: 1-D contiguous global->LDS DMA of `nelem` fp32 (nelem <= 65535).
// Builds a D# per CDNA5 ISA 8.3/8.4: count=1, data_size=4B, tile nelem x 1.
// Must be issued from a single wave; tracked by TENSORcnt.
// ---------------------------------------------------------------------------
__device__ inline void tdm_load_f32(const float* gsrc, float* ldst, int nelem)
{
  unsigned long long ga = (unsigned long long)(uintptr_t)gsrc;
  unsigned int laddr = (unsigned int)(uintptr_t)ldst;   // low 32 bits = LDS offset

  u32x4 g0;
  g0[0] = 1u;                                           // count=1, user desc
  g0[1] = laddr;                                        // lds_addr (bytes)
  g0[2] = (unsigned int)(ga & 0xffffffffu);             // global_addr[31:0]
  g0[3] = (unsigned int)((ga >> 32) & 0x01ffffffu)      // global_addr[56:32]
        | (2u << 30);                                   // type = 2 ("image")

  unsigned int ne = (unsigned int)nelem;
  i32x8 g1;
  g1[0] = (int)(2u << 16);                 // workgroup_mask=0, data_size=4B
  g1[1] = (int)((ne & 0xffffu) << 16);     // tensor_dim0[15:0] at bits 63:48
  g1[2] = (int)((ne >> 16) & 0xffffu)      // tensor_dim0[31:16] at bits 79:64
        | (1 << 16);                       // tensor_dim1 = 1 (bits 95:80)
  g1[3] = (int)((ne & 0xffffu) << 16);     // tile_dim0 = nelem (bits 127:112)
  g1[4] = 1;                               // tile_dim1 = 1, tile_dim2 = 0
  g1[5] = (int)ne;                         // tensor_dim0_stride[31:0]
  g1[6] = 0;
  g1[7] = 0;

  i32x4 z4 = {0, 0, 0, 0};
#if __clang_major__ >= 23
  i32x8 z8 = {0, 0, 0, 0, 0, 0, 0, 0};
  __builtin_amdgcn_tensor_load_to_lds(g0, g1, z4, z4, z8, 0);
#else
  __builtin_amdgcn_tensor_load_to_lds(g0, g1, z4, z4, 0);
#endif
}

// sum across the 4 adjacent lanes {base, base+1, base+2, base+3}
__device__ inline float red4(float v)
{
  v += __shfl_xor(v, 1);
  v += __shfl_xor(v, 2);
  return v;
}

// ---------------------------------------------------------------------------
// Generic fp32 WMMA GEMM: C[M,N] = A[M,K] @ B[K,N]   (TRANSB: B given as [N,K])
// Block tile 128x128, K-tile 16, 256 threads = 8 waves, wave tile 32x64.
// ---------------------------------------------------------------------------
template <bool TRANSB>
__global__ __launch_bounds__(256)
void gemm_f32_wmma(const float* __restrict__ A, const float* __restrict__ Bm,
                   float* __restrict__ C, int Mdim, int Ndim, int Kdim)
{
  constexpr int BM = 128, BN = 128, BK = 16;
  __shared__ float As[BM][BK + 1];
  __shared__ float Bs[BK][BN];

  const int t    = threadIdx.x;
  const int lane = t & 31;
  const int wv   = t >> 5;
  const int half = lane >> 4;
  const int lid  = lane & 15;
  const int wm   = wv >> 1;
  const int wn   = wv & 1;
  const int rowBase = blockIdx.y * BM;
  const int colBase = blockIdx.x * BN;

  v8f zero = {};
  v8f acc[2][4];
  for (int mi = 0; mi < 2; ++mi)
    for (int ni = 0; ni < 4; ++ni) acc[mi][ni] = zero;

  for (int k0 = 0; k0 < Kdim; k0 += BK) {
    for (int id = t; id < 512; id += 256) {
      int r = id >> 2, c4 = (id & 3) * 4;
      float4 v = *reinterpret_cast<const float4*>(
          A + (size_t)(rowBase + r) * Kdim + k0 + c4);
      As[r][c4 + 0] = v.x; As[r][c4 + 1] = v.y;
      As[r][c4 + 2] = v.z; As[r][c4 + 3] = v.w;
    }
    if (!TRANSB) {
      for (int id = t; id < 512; id += 256) {
        int r = id >> 5, c4 = (id & 31) * 4;
        int gn = colBase + c4;
        float4 v = make_float4(0.f, 0.f, 0.f, 0.f);
        if (gn + 3 < Ndim)
          v = *reinterpret_cast<const float4*>(Bm + (size_t)(k0 + r) * Ndim + gn);
        Bs[r][c4 + 0] = v.x; Bs[r][c4 + 1] = v.y;
        Bs[r][c4 + 2] = v.z; Bs[r][c4 + 3] = v.w;
      }
    } else {
      for (int id = t; id < 2048; id += 256) {
        int nn = id >> 4, kk = id & 15;
        int gn = colBase + nn;
        Bs[kk][nn] = (gn < Ndim) ? Bm[(size_t)gn * Kdim + k0 + kk] : 0.f;
      }
    }
    __syncthreads();

    for (int kk = 0; kk < BK; kk += 4) {
      const int kf = kk + 2 * half;
      v2f af[2], bf[4];
      for (int mi = 0; mi < 2; ++mi) {
        const int mr = wm * 32 + mi * 16 + lid;
        af[mi].x = As[mr][kf];
        af[mi].y = As[mr][kf + 1];
      }
      for (int ni = 0; ni < 4; ++ni) {
        const int nc = wn * 64 + ni * 16 + lid;
        bf[ni].x = Bs[kf][nc];
        bf[ni].y = Bs[kf + 1][nc];
      }
      for (int mi = 0; mi < 2; ++mi)
        for (int ni = 0; ni < 4; ++ni)
          acc[mi][ni] = WMMA_F32X4(af[mi], bf[ni], acc[mi][ni]);
    }
    __syncthreads();
  }

  for (int mi = 0; mi < 2; ++mi)
    for (int ni = 0; ni < 4; ++ni)
      for (int r = 0; r < 8; ++r) {
        const int gm = rowBase + wm * 32 + mi * 16 + r + 8 * half;
        const int gn = colBase + wn * 64 + ni * 16 + lid;
        if (gn < Ndim) C[(size_t)gm * Ndim + gn] = acc[mi][ni][r];
      }
}

// ---------------------------------------------------------------------------
// Fused: causal depthwise conv (q,k) -> head layout, XV -> head layout,
// sigmoid learning-rate, exact GELU gate.  One thread per (b, n, d).
// ---------------------------------------------------------------------------
__global__ __launch_bounds__(256)
void prep_kernel(const float* __restrict__ XQKV,
                 const float* __restrict__ cqw, const float* __restrict__ cqb,
                 const float* __restrict__ ckw, const float* __restrict__ ckb,
                 const float* __restrict__ lrb,
                 float* __restrict__ XQh, float* __restrict__ XKh,
                 float* __restrict__ XVh, float* __restrict__ lrh,
                 float* __restrict__ gate)
{
  const int idx = blockIdx.x * 256 + threadIdx.x;
  const int d = idx & (Dd - 1);
  const int n = (idx >> 10) & (Nn - 1);
  const int b = idx >> 22;
  const size_t rowb = (size_t)(b * Nn + n) * COLS;

  float q = cqb[d], k = ckb[d];
  for (int j = 0; j < Kc; ++j) {
    const int np = n - (Kc - 1) + j;
    if (np >= 0) {
      const float x = XQKV[(size_t)(b * Nn + np) * COLS + d];
      q += x * cqw[d * Kc + j];
      k += x * ckw[d * Kc + j];
    }
  }
  const int h = d >> 6, f = d & 63;
  const size_t ho = ((size_t)(b * NHh + h) * Nn + n) * HFf + f;
  XQh[ho] = q;
  XKh[ho] = k;
  XVh[ho] = XQKV[rowb + 2 * Dd + d];

  const float xg = XQKV[rowb + Dd + d];
  gate[(size_t)idx] = 0.5f * xg * (1.0f + erff(xg * 0.70710678118654752f));

  if (d < NHh) {
    const float lr_raw = XQKV[rowb + 3 * Dd + d];
    const float s = 1.0f / (1.0f + expf(-(lr_raw + lrb[d])));
    lrh[(size_t)(b * NHh + d) * Nn + n] = s * (1.0f / 64.0f);
  }
}

// ---------------------------------------------------------------------------
// Sequential TTT scan: one block per (b,h); W1 (64x64) resident in LDS.
// TDM double-buffers xq/xk/xv chunk loads; LN phases use 4 lanes per row
// with wave32 shfl reductions.
// LDS: 10*4096 + 256 floats = ~161 KB (within 320 KB / WGP).
// Ping-pong buffer pointers are computed arithmetically each iteration
// (no LDS-pointer arrays: those constant-fold into addrspacecast statics).
// ---------------------------------------------------------------------------
__global__ __launch_bounds__(256)
void ttt_scan_kernel(const float* __restrict__ XQh, const float* __restrict__ XKh,
                     const float* __restrict__ XVh, const float* __restrict__ lrh,
                     const float* __restrict__ W1g, const float* __restrict__ b1g,
                     const float* __restrict__ gw,  const float* __restrict__ gb,
                     float* __restrict__ outh)
{
  extern __shared__ float smem[];
  float* W1s    = smem;                   // 64x64 state
  // buffers: smem + (1+3*pp)*4096 = xq, (2+3*pp)*4096 = xk, (3+3*pp)*4096 = xv
  float* z1s    = smem + 7 * 4096;        // Z1 -> Z1hat -> Z1_bar
  float* gz1s   = smem + 8 * 4096;
  float* attns  = smem + 9 * 4096;
  float* b1s    = smem + 10 * 4096;       // 64
  float* etas   = b1s + 64;
  float* gammas = etas + 64;
  float* betas  = gammas + 64;

  const int bh   = blockIdx.x;
  const int h    = bh & (NHh - 1);
  const int t    = threadIdx.x;
  const int lane = t & 31;
  const int wv   = t >> 5;
  const int half = lane >> 4;
  const int lid  = lane & 15;
  const int row4 = t >> 2;                // 0..63 : row (or column) owner
  const int sub4 = t & 3;                 // 0..3  : 16-element segment
  const int fb   = sub4 * 16;

  for (int i = t; i < 4096; i += 256) W1s[i] = W1g[h * 4096 + i];
  if (t < 64) {
    b1s[t]    = b1g[h * 64 + t];
    gammas[t] = gw[h * 64 + t];
    betas[t]  = gb[h * 64 + t];
  }

  const size_t base = (size_t)bh * Nn * HFf;

  // prefetch chunk 0 via TDM (single wave issues; TENSORcnt tracks)
  if (wv == 0) {
    tdm_load_f32(XQh + base, smem + 1 * 4096, Mc * HFf);
    tdm_load_f32(XKh + base, smem + 2 * 4096, Mc * HFf);
    tdm_load_f32(XVh + base, smem + 3 * 4096, Mc * HFf);
  }

  v8f zero = {};

  for (int c = 0; c < Nn / Mc; ++c) {
    const int pp = c & 1;
    float* xqs = smem + (1 + 3 * pp) * 4096;
    float* xks = smem + (2 + 3 * pp) * 4096;
    float* xvs = smem + (3 + 3 * pp) * 4096;
    const size_t cb = base + (size_t)c * Mc * HFf;

    // issue next chunk's DMA, then wait for this chunk's 3 transfers
    if (wv == 0) {
      if (c + 1 < Nn / Mc) {
        const size_t nb = cb + (size_t)Mc * HFf;
        const int np = pp ^ 1;
        tdm_load_f32(XQh + nb, smem + (1 + 3 * np) * 4096, Mc * HFf);
        tdm_load_f32(XKh + nb, smem + (2 + 3 * np) * 4096, Mc * HFf);
        tdm_load_f32(XVh + nb, smem + (3 + 3 * np) * 4096, Mc * HFf);
        __builtin_amdgcn_s_wait_tensorcnt(3);
      } else {
        __builtin_amdgcn_s_wait_tensorcnt(0);
      }
    }
    if (t < 64) {
      const float e = lrh[(size_t)bh * Nn + c * Mc + t];
      etas[t] = fminf(fmaxf(e, 1e-6f), 1.0f);
    }
    __syncthreads();

    // ---- Z1 = xk @ W1 + b1 : 16 tiles, 2 per wave ----
    for (int tix = 0; tix < 2; ++tix) {
      const int ti = wv * 2 + tix;
      const int m0 = (ti >> 2) * 16, n0 = (ti & 3) * 16;
      v8f acc;
      const float binit = b1s[n0 + lid];
      for (int r = 0; r < 8; ++r) acc[r] = binit;
      for (int k = 0; k < 64; k += 4) {
        const int kf = k + 2 * half;
        v2f a, b;
        a.x = xks[(m0 + lid) * 64 + kf];  a.y = xks[(m0 + lid) * 64 + kf + 1];
        b.x = W1s[kf * 64 + n0 + lid];    b.y = W1s[(kf + 1) * 64 + n0 + lid];
        acc = WMMA_F32X4(a, b, acc);
      }
      for (int r = 0; r < 8; ++r)
        z1s[(m0 + r + 8 * half) * 64 + n0 + lid] = acc[r];
    }
    __syncthreads();

    // ---- LN(EPS_IN) + clipped gradient gZ1 : 4 lanes per row ----
    {
      const int m = row4;
      float* zr = z1s + m * 64;
      float p = 0.f;
      for (int f = 0; f < 16; ++f) p += zr[fb + f];
      const float mu = red4(p) * (1.0f / 64.0f);
      p = 0.f;
      for (int f = 0; f < 16; ++f) { const float d0 = zr[fb + f] - mu; p += d0 * d0; }
      const float var  = red4(p) * (1.0f / 64.0f);
      const float stdv = sqrtf(var + 1e-5f);
      const float inv  = 1.0f / stdv;
      float s1p = 0.f, s2p = 0.f;
      for (int f = 0; f < 16; ++f) {
        const int fi = fb + f;
        const float zh = (zr[fi] - mu) * inv;
        zr[fi] = zh;
        const float y = gammas[fi] * zh + betas[fi];
        float g = y - (xvs[m * 64 + fi] - xks[m * 64 + fi]);
        g = fminf(fmaxf(g, -10.f), 10.f);
        const float gxh = g * gammas[fi];
        gz1s[m * 64 + fi] = gxh;
        s1p += gxh; s2p += gxh * zh;
      }
      const float s1 = red4(s1p);
      const float s2 = red4(s2p);
      const float sc = 1.0f / (64.0f * stdv);
      for (int f = 0; f < 16; ++f) {
        const int fi = fb + f;
        const float gz = (64.f * gz1s[m * 64 + fi] - s1 - zr[fi] * s2) * sc;
        gz1s[m * 64 + fi] = fminf(fmaxf(gz, -5.f), 5.f);
      }
    }
    __syncthreads();

    // ---- Attn = tril(xq @ xk^T) * eta(row) ----
    for (int tix = 0; tix < 2; ++tix) {
      const int ti = wv * 2 + tix;
      const int m0 = (ti >> 2) * 16, n0 = (ti & 3) * 16;
      v8f acc = zero;
      for (int k = 0; k < 64; k += 4) {
        const int kf = k + 2 * half;
        v2f a, b;
        a.x = xqs[(m0 + lid) * 64 + kf];  a.y = xqs[(m0 + lid) * 64 + kf + 1];
        b.x = xks[(n0 + lid) * 64 + kf];  b.y = xks[(n0 + lid) * 64 + kf + 1];
        acc = WMMA_F32X4(a, b, acc);
      }
      for (int r = 0; r < 8; ++r) {
        const int mr = m0 + r + 8 * half, nc = n0 + lid;
        attns[mr * 64 + nc] = (nc <= mr) ? acc[r] * etas[mr] : 0.f;
      }
    }
    __syncthreads();

    // ---- b1_bar[m,d] = b1[d] - eta[m]*cumsum_m(gZ1) : 4-lane segmented scan ----
    {
      const int col = row4, seg = sub4;
      float sp = 0.f;
      for (int m = seg * 16; m < seg * 16 + 16; ++m) sp += gz1s[m * 64 + col];
      const int lb = lane & ~3;
      const float p0 = __shfl(sp, lb);
      const float p1 = __shfl(sp, lb + 1);
      const float p2 = __shfl(sp, lb + 2);
      float cs = (seg > 0 ? p0 : 0.f) + (seg > 1 ? p1 : 0.f) + (seg > 2 ? p2 : 0.f);
      const float bcol = b1s[col];
      for (int m = seg * 16; m < seg * 16 + 16; ++m) {
        cs += gz1s[m * 64 + col];
        xvs[m * 64 + col] = bcol - etas[m] * cs;
      }
    }
    __syncthreads();

    // ---- Z1_bar = xq @ W1 - Attn @ gZ1 + b1_bar ----
    for (int tix = 0; tix < 2; ++tix) {
      const int ti = wv * 2 + tix;
      const int m0 = (ti >> 2) * 16, n0 = (ti & 3) * 16;
      v8f acc;
      for (int r = 0; r < 8; ++r)
        acc[r] = xvs[(m0 + r + 8 * half) * 64 + n0 + lid];
      for (int k = 0; k < 64; k += 4) {
        const int kf = k + 2 * half;
        v2f a, b;
        a.x = xqs[(m0 + lid) * 64 + kf];  a.y = xqs[(m0 + lid) * 64 + kf + 1];
        b.x = W1s[kf * 64 + n0 + lid];    b.y = W1s[(kf + 1) * 64 + n0 + lid];
        acc = WMMA_F32X4(a, b, acc);
      }
      for (int k = 0; k < 64; k += 4) {
        const int kf = k + 2 * half;
        v2f a, b;
        a.x = attns[(m0 + lid) * 64 + kf]; a.y = attns[(m0 + lid) * 64 + kf + 1];
        b.x = -gz1s[kf * 64 + n0 + lid];   b.y = -gz1s[(kf + 1) * 64 + n0 + lid];
        acc = WMMA_F32X4(a, b, acc);
      }
      for (int r = 0; r < 8; ++r)
        z1s[(m0 + r + 8 * half) * 64 + n0 + lid] = acc[r];
    }
    __syncthreads();

    // ---- out = xq + gamma * LN(Z1_bar, EPS_OUT) + beta : 4 lanes per row ----
    {
      const int m = row4;
      float* zr = z1s + m * 64;
      float p = 0.f;
      for (int f = 0; f < 16; ++f) p += zr[fb + f];
      const float mu = red4(p) * (1.0f / 64.0f);
      p = 0.f;
      for (int f = 0; f < 16; ++f) { const float d0 = zr[fb + f] - mu; p += d0 * d0; }
      const float inv = 1.0f / sqrtf(red4(p) * (1.0f / 64.0f) + 1e-6f);
      float4 o[4];
      float* ov = reinterpret_cast<float*>(o);
      for (int f = 0; f < 16; ++f) {
        const int fi = fb + f;
        ov[f] = xqs[m * 64 + fi] + gammas[fi] * ((zr[fi] - mu) * inv) + betas[fi];
      }
      float4* og = reinterpret_cast<float4*>(outh + cb + m * 64 + fb);
      for (int f4 = 0; f4 < 4; ++f4) og[f4] = o[f4];
    }
    __syncthreads();

    // ---- state update: W1 -= last_eta * xk^T @ gZ1 ; b1 -= last_eta*colsum ----
    const float leta = etas[63];
    for (int tix = 0; tix < 2; ++tix) {
      const int ti = wv * 2 + tix;
      const int m0 = (ti >> 2) * 16, n0 = (ti & 3) * 16;
      v8f acc = zero;
      for (int k = 0; k < 64; k += 4) {
        const int kf = k + 2 * half;
        v2f a, b;
        a.x = xks[kf * 64 + m0 + lid];   a.y = xks[(kf + 1) * 64 + m0 + lid];
        b.x = gz1s[kf * 64 + n0 + lid];  b.y = gz1s[(kf + 1) * 64 + n0 + lid];
        acc = WMMA_F32X4(a, b, acc);
      }
      for (int r = 0; r < 8; ++r) {
        const int idx2 = (m0 + r + 8 * half) * 64 + n0 + lid;
        W1s[idx2] -= leta * acc[r];
      }
    }
    {
      const int col = row4, seg = sub4;
      float sp = 0.f;
      for (int m = seg * 16; m < seg * 16 + 16; ++m) sp += gz1s[m * 64 + col];
      const float s = red4(sp);
      if (seg == 0) b1s[col] -= leta * s;
    }
    __syncthreads();
  }
}

// ---------------------------------------------------------------------------
// Gather heads -> token row, LN over D=1024 (EPS_OUT), post-norm affine, gate.
// One block (256 threads) per token.
// ---------------------------------------------------------------------------
__global__ __launch_bounds__(256)
void merge_ln_gate_kernel(const float* __restrict__ outh, const float* __restrict__ gate,
                          const float* __restrict__ pw, const float* __restrict__ pb,
                          float* __restrict__ y)
{
  __shared__ float red[256];
  const int tok = blockIdx.x;
  const int b = tok >> 12, n = tok & 4095;
  const int t = threadIdx.x;

  float v[4];
  float s = 0.f;
  for (int i = 0; i < 4; ++i) {
    const int d = t + i * 256;
    const int h = d >> 6, f = d & 63;
    v[i] = outh[((size_t)(b * NHh + h) * Nn + n) * HFf + f];
    s += v[i];
  }
  red[t] = s; __syncthreads();
  for (int off = 128; off > 0; off >>= 1) {
    if (t < off) red[t] += red[t + off];
    __syncthreads();
  }
  const float mu = red[0] * (1.0f / 1024.0f);
  __syncthreads();

  s = 0.f;
  for (int i = 0; i < 4; ++i) { const float d0 = v[i] - mu; s += d0 * d0; }
  red[t] = s; __syncthreads();
  for (int off = 128; off > 0; off >>= 1) {
    if (t < off) red[t] += red[t + off];
    __syncthreads();
  }
  const float inv = 1.0f / sqrtf(red[0] * (1.0f / 1024.0f) + 1e-6f);

  for (int i = 0; i < 4; ++i) {
    const int d = t + i * 256;
    const float nh = (v[i] - mu) * inv;
    y[(size_t)tok * Dd + d] = gate[(size_t)tok * Dd + d] * (nh * pw[d] + pb[d]);
  }
}

// ---------------------------------------------------------------------------
extern "C" void kernel_launch(void* const* d_in, const int* in_sizes, int n_in,
                              void* d_out, int out_size, void* d_ws, size_t ws_size,
                              hipStream_t stream)
{
  const float* hs   = (const float*)d_in[0];
  const float* wqkv = (const float*)d_in[1];
  const float* lrb  = (const float*)d_in[2];
  const float* W1   = (const float*)d_in[3];
  const float* b1   = (const float*)d_in[4];
  const float* tnw  = (const float*)d_in[5];
  const float* tnb  = (const float*)d_in[6];
  const float* pnw  = (const float*)d_in[7];
  const float* pnb  = (const float*)d_in[8];
  const float* cqw  = (const float*)d_in[9];
  const float* cqb  = (const float*)d_in[10];
  const float* ckw  = (const float*)d_in[11];
  const float* ckb  = (const float*)d_in[12];
  const float* wo   = (const float*)d_in[13];

  constexpr size_t TOK = (size_t)Bb * Nn;          // 16384
  float* ws = (float*)d_ws;
  float* xqkvlr = ws;                               // TOK x 3088 (reused as y)
  size_t o = TOK * COLS;
  float* xqh  = ws + o; o += TOK * Dd;
  float* xkh  = ws + o; o += TOK * Dd;
  float* xvh  = ws + o; o += TOK * Dd;
  float* lrh  = ws + o; o += (size_t)Bb * NHh * Nn;
  float* gate = ws + o; o += TOK * Dd;
  float* outh = ws + o; o += TOK * Dd;
  float* ybuf = xqkvlr;

  dim3 blk(256);

  // 1) XQKV_lr = hs @ w_qkvlr   (16384 x 3088 x 1024)
  {
    dim3 grid((COLS + 127) / 128, (int)(TOK / 128));
    gemm_f32_wmma<false><<<grid, blk, 0, stream>>>(hs, wqkv, xqkvlr,
                                                   (int)TOK, COLS, Dd);
  }
  // 2) conv q/k + head scatter + lr + gate
  prep_kernel<<<dim3((unsigned)(TOK * Dd / 256)), blk, 0, stream>>>(
      xqkvlr, cqw, cqb, ckw, ckb, lrb, xqh, xkh, xvh, lrh, gate);

  // 3) sequential TTT scan, one block per (b,h); TDM double-buffered feeds
  {
    const size_t shbytes = (10 * 4096 + 4 * 64) * sizeof(float);
    ttt_scan_kernel<<<dim3(Bb * NHh), blk, shbytes, stream>>>(
        xqh, xkh, xvh, lrh, W1, b1, tnw, tnb, outh);
  }
  // 4a) merge heads + post-LN + gate -> y
  merge_ln_gate_kernel<<<dim3((unsigned)TOK), blk, 0, stream>>>(
      outh, gate, pnw, pnb, ybuf);

  // 4b) out = y @ wo^T   (16384 x 1024 x 1024)
  {
    dim3 grid(Dd / 128, (int)(TOK / 128));
    gemm_f32_wmma<true><<<grid, blk, 0, stream>>>(ybuf, wo, (float*)d_out,
                                                  (int)TOK, Dd, Dd);
  }
}